// GIN_21260088115449
// MI455X (gfx1250) — compile-verified
//
#include <hip/hip_runtime.h>

typedef __attribute__((ext_vector_type(2))) float v2f;
typedef __attribute__((ext_vector_type(8))) float v8f;

#define ST 68  // LDS staging row stride (floats) to break bank alignment

__device__ inline v8f wmma_k4(v2f a, v2f b, v8f c) {
  // D = A(16x4,f32) * B(4x16,f32) + C(16x16,f32)
  return __builtin_amdgcn_wmma_f32_16x16x4_f32(
      /*neg_a=*/false, a, /*neg_b=*/false, b,
      /*c_mod=*/(short)0, c, /*reuse_a=*/false, /*reuse_b=*/false);
}

__device__ inline v8f splat8(float x) {
  v8f v = {x, x, x, x, x, x, x, x};
  return v;
}

// ---------------------------------------------------------------- copy ----
__global__ void copy_vec4(const float4* __restrict__ s, float4* __restrict__ d, int n4) {
  int i = blockIdx.x * blockDim.x + threadIdx.x;
  int stride = gridDim.x * blockDim.x;
  for (; i < n4; i += stride) d[i] = s[i];
}

// ------------------------------------------------------------- scatter ----
// One wave (32 lanes) per edge; each lane handles C/32 channels with
// float atomics (global_atomic_add_f32, no return).
template <int C>
__global__ void scatter_add(const float* __restrict__ feat, float* __restrict__ agg,
                            const long long* __restrict__ sidx,
                            const long long* __restrict__ didx, int n_edges) {
  constexpr int PER = C / 32;
  long long gid = (long long)blockIdx.x * blockDim.x + threadIdx.x;
  int e = (int)(gid >> 5);
  int lane = (int)(gid & 31);
  if (e >= n_edges) return;
  int s = (int)sidx[e];
  int d = (int)didx[e];
  const float* fs = feat + (long long)s * C + lane * PER;
  float* fd = agg + (long long)d * C + lane * PER;
  if constexpr (PER == 4) {
    float4 v = *(const float4*)fs;
    atomicAdd(fd + 0, v.x);
    atomicAdd(fd + 1, v.y);
    atomicAdd(fd + 2, v.z);
    atomicAdd(fd + 3, v.w);
  } else {
    float2 v = *(const float2*)fs;
    atomicAdd(fd + 0, v.x);
    atomicAdd(fd + 1, v.y);
  }
}

// ---------------------------------------------------- GIN MLP 1 (128->64->64)
// 64 threads = 2 waves per block; each wave computes a 16-node tile.
// out = relu( relu(A@W1 + b1) @ W2 + b2 ), A is [n][128].
__global__ void __launch_bounds__(64)
gin_mlp1(const float* __restrict__ A, const float* __restrict__ W1,
         const float* __restrict__ b1, const float* __restrict__ W2,
         const float* __restrict__ b2, float* __restrict__ out) {
  __shared__ float lW1[128 * 64];
  __shared__ float lW2[64 * 64];
  __shared__ float lb1[64], lb2[64];
  __shared__ float stage[2][16 * ST];

  int tid = threadIdx.x;
  for (int i = tid; i < 128 * 64; i += 64) lW1[i] = W1[i];
  for (int i = tid; i < 64 * 64; i += 64) lW2[i] = W2[i];
  if (tid < 64) { lb1[tid] = b1[tid]; lb2[tid] = b2[tid]; }
  __syncthreads();

  int wave = tid >> 5;
  int lane = tid & 31;
  int row = lane & 15;   // M index (A) / N index (B,C)
  int kh = lane >> 4;    // K half-select
  int node0 = (blockIdx.x * 2 + wave) * 16;
  float* st = stage[wave];

  // ---- layer 1: h1 = relu(A@W1 + b1), K=128 ----
  v8f acc[4];
#pragma unroll
  for (int t = 0; t < 4; ++t) acc[t] = splat8(lb1[t * 16 + row]);

  const float* arow = A + (long long)(node0 + row) * 128 + kh * 2;
#pragma unroll 4
  for (int ks = 0; ks < 128; ks += 4) {
    v2f a;
    a.x = arow[ks];
    a.y = arow[ks + 1];
#pragma unroll
    for (int t = 0; t < 4; ++t) {
      v2f b;
      b.x = lW1[(ks + kh * 2) * 64 + t * 16 + row];
      b.y = lW1[(ks + kh * 2 + 1) * 64 + t * 16 + row];
      acc[t] = wmma_k4(a, b, acc[t]);
    }
  }
  // relu + stage to LDS (C-layout -> [M][N] tile)
#pragma unroll
  for (int t = 0; t < 4; ++t)
#pragma unroll
    for (int v = 0; v < 8; ++v) {
      float z = acc[t][v];
      st[(v + kh * 8) * ST + t * 16 + row] = z > 0.f ? z : 0.f;
    }

  // ---- layer 2: out = relu(h1@W2 + b2), K=64 ----
  v8f acc2[4];
#pragma unroll
  for (int t = 0; t < 4; ++t) acc2[t] = splat8(lb2[t * 16 + row]);
#pragma unroll 4
  for (int ks = 0; ks < 64; ks += 4) {
    v2f a;
    a.x = st[row * ST + ks + kh * 2];
    a.y = st[row * ST + ks + kh * 2 + 1];
#pragma unroll
    for (int t = 0; t < 4; ++t) {
      v2f b;
      b.x = lW2[(ks + kh * 2) * 64 + t * 16 + row];
      b.y = lW2[(ks + kh * 2 + 1) * 64 + t * 16 + row];
      acc2[t] = wmma_k4(a, b, acc2[t]);
    }
  }
#pragma unroll
  for (int t = 0; t < 4; ++t)
#pragma unroll
    for (int v = 0; v < 8; ++v) {
      float z = acc2[t][v];
      out[((long long)node0 + v + kh * 8) * 64 + t * 16 + row] = z > 0.f ? z : 0.f;
    }
}

// -------------------------------- GIN MLP 2 + final FC + log_softmax ------
// out = log_softmax( relu(relu(A@W3+b3)@W4+b4) @ Wf + bf ), A is [n][64].
__global__ void __launch_bounds__(64)
gin_mlp2_head(const float* __restrict__ A, const float* __restrict__ W3,
              const float* __restrict__ b3, const float* __restrict__ W4,
              const float* __restrict__ b4, const float* __restrict__ Wf,
              const float* __restrict__ bf, float* __restrict__ out) {
  __shared__ float lW3[64 * 64];
  __shared__ float lW4[64 * 64];
  __shared__ float lWf[64 * 48];  // Wf zero-padded 40 -> 48 cols
  __shared__ float lb3[64], lb4[64], lbf[48];
  __shared__ float stage[2][16 * ST];

  int tid = threadIdx.x;
  for (int i = tid; i < 64 * 64; i += 64) { lW3[i] = W3[i]; lW4[i] = W4[i]; }
  for (int i = tid; i < 64 * 48; i += 64) {
    int r = i / 48, c = i % 48;
    lWf[i] = (c < 40) ? Wf[r * 40 + c] : 0.f;
  }
  if (tid < 64) { lb3[tid] = b3[tid]; lb4[tid] = b4[tid]; }
  if (tid < 48) lbf[tid] = (tid < 40) ? bf[tid] : 0.f;
  __syncthreads();

  int wave = tid >> 5;
  int lane = tid & 31;
  int row = lane & 15;
  int kh = lane >> 4;
  int node0 = (blockIdx.x * 2 + wave) * 16;
  float* st = stage[wave];

  // ---- layer 3: h1 = relu(A@W3 + b3), K=64 ----
  v8f acc[4];
#pragma unroll
  for (int t = 0; t < 4; ++t) acc[t] = splat8(lb3[t * 16 + row]);
  const float* arow = A + (long long)(node0 + row) * 64 + kh * 2;
#pragma unroll 4
  for (int ks = 0; ks < 64; ks += 4) {
    v2f a;
    a.x = arow[ks];
    a.y = arow[ks + 1];
#pragma unroll
    for (int t = 0; t < 4; ++t) {
      v2f b;
      b.x = lW3[(ks + kh * 2) * 64 + t * 16 + row];
      b.y = lW3[(ks + kh * 2 + 1) * 64 + t * 16 + row];
      acc[t] = wmma_k4(a, b, acc[t]);
    }
  }
#pragma unroll
  for (int t = 0; t < 4; ++t)
#pragma unroll
    for (int v = 0; v < 8; ++v) {
      float z = acc[t][v];
      st[(v + kh * 8) * ST + t * 16 + row] = z > 0.f ? z : 0.f;
    }

  // ---- layer 4: h2 = relu(h1@W4 + b4), K=64 (h2 overwrites stage) ----
  v8f acc2[4];
#pragma unroll
  for (int t = 0; t < 4; ++t) acc2[t] = splat8(lb4[t * 16 + row]);
#pragma unroll 4
  for (int ks = 0; ks < 64; ks += 4) {
    v2f a;
    a.x = st[row * ST + ks + kh * 2];
    a.y = st[row * ST + ks + kh * 2 + 1];
#pragma unroll
    for (int t = 0; t < 4; ++t) {
      v2f b;
      b.x = lW4[(ks + kh * 2) * 64 + t * 16 + row];
      b.y = lW4[(ks + kh * 2 + 1) * 64 + t * 16 + row];
      acc2[t] = wmma_k4(a, b, acc2[t]);
    }
  }
  v8f h2[4];
#pragma unroll
  for (int t = 0; t < 4; ++t) h2[t] = acc2[t];
  // stash h2 into stage (all h1 reads already done in program order)
#pragma unroll
  for (int t = 0; t < 4; ++t)
#pragma unroll
    for (int v = 0; v < 8; ++v) {
      float z = h2[t][v];
      st[(v + kh * 8) * ST + t * 16 + row] = z > 0.f ? z : 0.f;
    }

  // ---- final FC: logits = h2@Wf + bf, K=64, N=48 (40 valid) ----
  v8f accF[3];
#pragma unroll
  for (int t = 0; t < 3; ++t) accF[t] = splat8(lbf[t * 16 + row]);
#pragma unroll 4
  for (int ks = 0; ks < 64; ks += 4) {
    v2f a;
    a.x = st[row * ST + ks + kh * 2];
    a.y = st[row * ST + ks + kh * 2 + 1];
#pragma unroll
    for (int t = 0; t < 3; ++t) {
      v2f b;
      b.x = lWf[(ks + kh * 2) * 48 + t * 16 + row];
      b.y = lWf[(ks + kh * 2 + 1) * 48 + t * 16 + row];
      accF[t] = wmma_k4(a, b, accF[t]);
    }
  }
  // logits into stage, then row-wise log_softmax by lanes 0..15
#pragma unroll
  for (int t = 0; t < 3; ++t)
#pragma unroll
    for (int v = 0; v < 8; ++v)
      st[(v + kh * 8) * ST + t * 16 + row] = accF[t][v];

  if (lane < 16) {
    const float* lrow = &st[lane * ST];
    float m = -3.402823466e38f;
    for (int j = 0; j < 40; ++j) m = fmaxf(m, lrow[j]);
    float ssum = 0.f;
    for (int j = 0; j < 40; ++j) ssum += __expf(lrow[j] - m);
    float lg = __logf(ssum);
    float* o = out + (long long)(node0 + lane) * 40;
    for (int j = 0; j < 40; ++j) o[j] = lrow[j] - m - lg;
  }
}

// ------------------------------------------------------------------ host --
extern "C" void kernel_launch(void* const* d_in, const int* in_sizes, int n_in,
                              void* d_out, int out_size, void* d_ws, size_t ws_size,
                              hipStream_t stream) {
  const float* x = (const float*)d_in[0];
  const long long* ei = (const long long*)d_in[1];  // int64 (2, E)
  const float* W1 = (const float*)d_in[2];
  const float* b1 = (const float*)d_in[3];
  const float* W2 = (const float*)d_in[4];
  const float* b2 = (const float*)d_in[5];
  const float* W3 = (const float*)d_in[6];
  const float* b3 = (const float*)d_in[7];
  const float* W4 = (const float*)d_in[8];
  const float* b4 = (const float*)d_in[9];
  const float* Wf = (const float*)d_in[10];
  const float* bf = (const float*)d_in[11];
  float* out = (float*)d_out;

  int n_nodes = in_sizes[0] / 128;  // 100000
  int n_edges = in_sizes[1] / 2;    // 3200000
  const long long* src = ei;
  const long long* dst = ei + n_edges;

  // workspace layout: [agg1: n*128 f32][hmid: n*64 f32]; agg2 aliases agg1
  float* agg1 = (float*)d_ws;
  float* hmid = agg1 + (size_t)n_nodes * 128;
  float* agg2 = agg1;

  // 1) agg1 = x  (h = x + sum_neighbors, eps = 0)
  int n4 = n_nodes * 128 / 4;
  copy_vec4<<<(n4 + 255) / 256, 256, 0, stream>>>((const float4*)x, (float4*)agg1, n4);

  // 2) scatter-add x[src] into agg1[dst]
  long long tthreads = (long long)n_edges * 32;
  int sblocks = (int)((tthreads + 255) / 256);
  scatter_add<128><<<sblocks, 256, 0, stream>>>(x, agg1, src, dst, n_edges);

  // 3) hmid = relu(MLP1(agg1))
  int tiles = n_nodes / 16;  // 6250
  gin_mlp1<<<tiles / 2, 64, 0, stream>>>(agg1, W1, b1, W2, b2, hmid);

  // 4) agg2 = hmid
  int n4b = n_nodes * 64 / 4;
  copy_vec4<<<(n4b + 255) / 256, 256, 0, stream>>>((const float4*)hmid, (float4*)agg2, n4b);

  // 5) scatter-add hmid[src] into agg2[dst]
  scatter_add<64><<<sblocks, 256, 0, stream>>>(hmid, agg2, src, dst, n_edges);

  // 6) out = log_softmax(MLP2(agg2) @ Wf + bf)
  gin_mlp2_head<<<tiles / 2, 64, 0, stream>>>(agg2, W3, b3, W4, b4, Wf, bf, out);
}